// MemoryRamModule_79697413145069
// MI455X (gfx1250) — compile-verified
//
#include <hip/hip_runtime.h>
#include <hip/hip_bf16.h>

// Sizes from the reference: I=1024, H=512, M=100, n_img=4096 (derived from in_sizes).
#define DI 1024
#define DH 512
#define DM 100

typedef __attribute__((ext_vector_type(2))) float v2f;
typedef __attribute__((ext_vector_type(8))) float v8f;

// ---------------------------------------------------------------------------
// Phase 1: C[M x N] = A[M x K] @ B[K x N]  (row-major, fp32) via
// V_WMMA_F32_16X16X4_F32.  One 16x16 C tile per wave, 8 waves per block
// (block covers 128 rows x 16 cols).  K unrolled by 8 (2 WMMAs / trip).
//
// fp32 WMMA fragment layouts (CDNA5 ISA):
//   A 16x4 : lane l (l<16): M=l, VGPR0=K0,VGPR1=K1 ; lanes 16-31: K2,K3
//   B 4x16 : lane l: N=l&15 ; VGPR0 holds K = 2*(l>>4), VGPR1 = K+1
//   C 16x16: vgpr g, lane l: row = g + 8*(l>>4), col = l&15
//
// GUARD=false for N % 16 == 0 launches (no column masking at all).
// GUARD=true clamps OOB columns to N-1 (legal address) and scales the
// loaded value by 0 -> branch-free, no exec-mask manipulation.
// ---------------------------------------------------------------------------
template <bool GUARD>
__global__ __launch_bounds__(256)
void wmma_gemm_f32(const float* __restrict__ A, const float* __restrict__ B,
                   float* __restrict__ C, int Mrows, int K, int N,
                   int lda, int ldb, int ldc)
{
    const int lane = threadIdx.x & 31;
    const int wave = threadIdx.x >> 5;
    const int row0 = (blockIdx.x * 8 + wave) * 16;
    const int col0 = blockIdx.y * 16;
    if (row0 >= Mrows) return;

    const int m   = lane & 15;            // A row within tile
    const int kk  = (lane >> 4) << 1;     // 0 or 2 (K sub-offset for this lane half)
    const int col = col0 + (lane & 15);   // B / C column
    const bool cok  = (!GUARD) || (col < N);
    const float bs  = cok ? 1.f : 0.f;    // zero contribution of padded columns
    const int  colc = cok ? col : (N - 1);

    // Hoisted, strength-reduced pointers (kk folded in).
    const float* aptr = A + (size_t)(row0 + m) * lda + kk;
    const float* bptr = B + (size_t)kk * ldb + colc;
    const size_t ldbz = (size_t)ldb;

    v8f acc = {0.f, 0.f, 0.f, 0.f, 0.f, 0.f, 0.f, 0.f};

    for (int k0 = 0; k0 < K; k0 += 8) {
        // A fragments: K slices [k0+kk, k0+kk+1] and [k0+4+kk, k0+4+kk+1]
        const v2f a0 = *(const v2f*)(aptr);
        const v2f a1 = *(const v2f*)(aptr + 4);
        // B fragments: rows k0+kk, k0+kk+1, k0+4+kk, k0+4+kk+1 (coalesced in lanes)
        v2f b0, b1;
        b0.x = bptr[0]        * bs;
        b0.y = bptr[ldbz]     * bs;
        b1.x = bptr[4 * ldbz] * bs;
        b1.y = bptr[5 * ldbz] * bs;
        acc = __builtin_amdgcn_wmma_f32_16x16x4_f32(
                  false, a0, false, b0, (short)0, acc, false, false);
        acc = __builtin_amdgcn_wmma_f32_16x16x4_f32(
                  false, a1, false, b1, (short)0, acc, false, false);
        aptr += 8;
        bptr += 8 * ldbz;
    }

#pragma unroll
    for (int g = 0; g < 8; ++g) {
        const int row = row0 + g + ((lane >> 4) << 3);
        if (cok) C[(size_t)row * ldc + col] = acc[g];
    }
}

// ---------------------------------------------------------------------------
// Phase 1b: scalar-gate precompute  P_wg[t] = X[t] . W_wg[:I],  same for rg.
// One block per frame, 256-thread LDS tree reduction.
// ---------------------------------------------------------------------------
__global__ __launch_bounds__(256)
void gates_kernel(const float* __restrict__ X, const float* __restrict__ W_wg,
                  const float* __restrict__ W_rg, float* __restrict__ P_wg,
                  float* __restrict__ P_rg)
{
    __shared__ float s1[256], s2[256];
    const int t = blockIdx.x, tid = threadIdx.x;
    float a = 0.f, b = 0.f;
    for (int k = tid; k < DI; k += 256) {
        const float x = X[(size_t)t * DI + k];
        a += x * W_wg[k];
        b += x * W_rg[k];
    }
    s1[tid] = a; s2[tid] = b;
    __syncthreads();
    for (int s = 128; s > 0; s >>= 1) {
        if (tid < s) { s1[tid] += s1[tid + s]; s2[tid] += s2[tid + s]; }
        __syncthreads();
    }
    if (tid == 0) { P_wg[t] = s1[0]; P_rg[t] = s2[0]; }
}

// ---------------------------------------------------------------------------
// Phase 2: sequential scan.  One persistent 1024-thread workgroup.
// mem (100x512 fp32 = 200KB) lives in LDS for the whole scan (320KB/WGP),
// avoiding ~1.6 GB of HBM traffic over 4096 steps.  Recurrent weights
// (~3.4MB/step) stream from the 192MB L2 (fully resident).
// ---------------------------------------------------------------------------
__global__ __launch_bounds__(1024, 1)
void scan_kernel(const float* __restrict__ P_xh, const float* __restrict__ P_hc,
                 const float* __restrict__ P_wp, const float* __restrict__ P_rp,
                 const float* __restrict__ P_wg, const float* __restrict__ P_rg,
                 const float* __restrict__ W_hc, const float* __restrict__ b_hc,
                 const float* __restrict__ W_wg, const float* __restrict__ b_wg,
                 const float* __restrict__ W_wp, const float* __restrict__ b_wp,
                 const float* __restrict__ W_rg, const float* __restrict__ b_rg,
                 const float* __restrict__ W_rp, const float* __restrict__ b_rp,
                 const float* __restrict__ Wrh, const float* __restrict__ Whh,
                 const float* __restrict__ bh, float* __restrict__ out, int nImg)
{
    extern __shared__ float smem[];
    float* s_mem  = smem;                 // 100*512
    float* s_h    = s_mem + DM * DH;      // 512
    float* s_r    = s_h + DH;             // 512
    float* s_c    = s_r + DH;             // 512
    float* s_ar   = s_c + DH;             // 128
    float* s_aw   = s_ar + 128;           // 128
    float* s_red  = s_aw + 128;           // 128
    float* s_p0   = s_red + 128;          // 1024
    float* s_p1   = s_p0 + 1024;          // 1024
    float* s_scal = s_p1 + 1024;          // 8: [0]max [1]1/sum [2]go [3]gw [4]max2 [5]1/sum2

    const int tid = threadIdx.x;

    // init state: h = 0, mem = 0
    for (int i = tid; i < DM * DH; i += 1024) s_mem[i] = 0.f;
    if (tid < DH) s_h[tid] = 0.f;
    __syncthreads();

    for (int t = 0; t < nImg; ++t) {
        // ---- P1: e_rp[i] = P_rp[t,i] + b_rp[i] + h . W_rp[I:, i]   (100 outs)
        {
            const int grp = tid >> 3, g = tid & 7;      // 8 lanes per output
            float p = 0.f;
            if (grp < DM)
                for (int k = g; k < DH; k += 8)
                    p += s_h[k] * W_rp[(size_t)(DI + k) * DM + grp];
            s_p0[tid] = p;
        }
        __syncthreads();
        if (tid < 128) {
            float v = -3.0e38f;
            if (tid < DM) {
                float acc = 0.f;
#pragma unroll
                for (int g = 0; g < 8; ++g) acc += s_p0[(tid << 3) + g];
                v = acc + P_rp[(size_t)t * DM + tid] + b_rp[tid];
            }
            s_red[tid] = v;
        }
        __syncthreads();
        // ---- P2: ar = softmax(e_rp)
        if (tid < 32)
            s_p0[tid] = fmaxf(fmaxf(s_red[tid], s_red[tid + 32]),
                              fmaxf(s_red[tid + 64], s_red[tid + 96]));
        __syncthreads();
        if (tid == 0) { float m = s_p0[0]; for (int i = 1; i < 32; ++i) m = fmaxf(m, s_p0[i]); s_scal[0] = m; }
        __syncthreads();
        if (tid < 128) s_ar[tid] = (tid < DM) ? __expf(s_red[tid] - s_scal[0]) : 0.f;
        __syncthreads();
        if (tid < 32) s_p0[tid] = s_ar[tid] + s_ar[tid + 32] + s_ar[tid + 64] + s_ar[tid + 96];
        __syncthreads();
        if (tid == 0) { float s = 0.f; for (int i = 0; i < 32; ++i) s += s_p0[i]; s_scal[1] = 1.f / s; }
        __syncthreads();
        if (tid < DM) s_ar[tid] *= s_scal[1];

        // ---- P3: gates  rg = h.W_rg[I:] (lanes 0-511), wg = h.W_wg[I:] (512-1023)
        {
            float p;
            if (tid < DH) p = s_h[tid] * W_rg[DI + tid];
            else          p = s_h[tid - DH] * W_wg[DI + (tid - DH)];
            s_p0[tid] = p;
        }
        __syncthreads();
        if (tid < 64) {
            float a = 0.f; for (int k = tid; k < DH; k += 64) a += s_p0[k];
            s_p1[tid] = a;
            float b = 0.f; for (int k = DH + tid; k < 2 * DH; k += 64) b += s_p0[k];
            s_p1[64 + tid] = b;
        }
        __syncthreads();
        if (tid == 0) {
            float rg = 0.f, wg = 0.f;
            for (int i = 0; i < 64; ++i) { rg += s_p1[i]; wg += s_p1[64 + i]; }
            rg += P_rg[t] + b_rg[0];
            wg += P_wg[t] + b_wg[0];
            s_scal[2] = 1.f / (1.f + __expf(-rg));   // go
            s_scal[3] = 1.f / (1.f + __expf(-wg));   // gw
        }
        __syncthreads();

        // ---- P4: aw = softmax(P_wp[t] + b_wp + h . W_wp[I:, :])
        {
            const int grp = tid >> 3, g = tid & 7;
            float p = 0.f;
            if (grp < DM)
                for (int k = g; k < DH; k += 8)
                    p += s_h[k] * W_wp[(size_t)(DI + k) * DM + grp];
            s_p0[tid] = p;
        }
        __syncthreads();
        if (tid < 128) {
            float v = -3.0e38f;
            if (tid < DM) {
                float acc = 0.f;
#pragma unroll
                for (int g = 0; g < 8; ++g) acc += s_p0[(tid << 3) + g];
                v = acc + P_wp[(size_t)t * DM + tid] + b_wp[tid];
            }
            s_red[tid] = v;
        }
        __syncthreads();
        if (tid < 32)
            s_p0[tid] = fmaxf(fmaxf(s_red[tid], s_red[tid + 32]),
                              fmaxf(s_red[tid + 64], s_red[tid + 96]));
        __syncthreads();
        if (tid == 0) { float m = s_p0[0]; for (int i = 1; i < 32; ++i) m = fmaxf(m, s_p0[i]); s_scal[4] = m; }
        __syncthreads();
        if (tid < 128) s_aw[tid] = (tid < DM) ? __expf(s_red[tid] - s_scal[4]) : 0.f;
        __syncthreads();
        if (tid < 32) s_p0[tid] = s_aw[tid] + s_aw[tid + 32] + s_aw[tid + 64] + s_aw[tid + 96];
        __syncthreads();
        if (tid == 0) { float s = 0.f; for (int i = 0; i < 32; ++i) s += s_p0[i]; s_scal[5] = 1.f / s; }
        __syncthreads();
        if (tid < DM) s_aw[tid] *= s_scal[5];
        __syncthreads();

        // ---- P5: r[j] = go * sum_i ar[i] * mem[i][j]   (i split over 2 halves)
        {
            const int j = tid & 511, half = tid >> 9;
            const int i0 = half * 50;
            float acc = 0.f;
            for (int i = i0; i < i0 + 50; ++i) acc += s_ar[i] * s_mem[i * DH + j];
            s_p0[tid] = acc;
        }
        __syncthreads();
        if (tid < DH) s_r[tid] = s_scal[2] * (s_p0[tid] + s_p0[DH + tid]);
        __syncthreads();

        // ---- P6: z = r@Wrh + h@Whh (+P_xh+bh), c_pre = h@W_hc[I:] (+P_hc+b_hc)
        {
            const int j = tid & 511, half = tid >> 9;
            float accz = 0.f;
            if (half == 0) { for (int k = 0; k < DH; ++k) accz += s_r[k] * Wrh[(size_t)k * DH + j]; }
            else           { for (int k = 0; k < DH; ++k) accz += s_h[k] * Whh[(size_t)k * DH + j]; }
            s_p0[tid] = accz;
            float accc = 0.f;
            const int k0 = half * 256;
            for (int k = k0; k < k0 + 256; ++k) accc += s_h[k] * W_hc[(size_t)(DI + k) * DH + j];
            s_p1[tid] = accc;
        }
        __syncthreads();
        if (tid < DH) {
            const float c  = fmaxf(s_p1[tid] + s_p1[DH + tid] + P_hc[(size_t)t * DH + tid] + b_hc[tid], 0.f);
            s_c[tid] = c;
            const float hn = fmaxf(s_p0[tid] + s_p0[DH + tid] + P_xh[(size_t)t * DH + tid] + bh[tid], 0.f);
            s_h[tid] = hn;                       // all reads of old h completed above
            out[(size_t)t * DH + tid] = hn;
        }
        __syncthreads();

        // ---- P7: mem = gw*aw[i]*c[j] + (1-aw[i])*mem   (50 elems per thread)
        {
            const float gw = s_scal[3];
            for (int e = 0; e < 50; ++e) {
                const int id = e * 1024 + tid;
                const int i = id >> 9, j = id & 511;
                const float a = s_aw[i];
                s_mem[id] = gw * a * s_c[j] + (1.f - a) * s_mem[id];
            }
        }
        __syncthreads();
    }
}

// ---------------------------------------------------------------------------
// Host launch
// ---------------------------------------------------------------------------
extern "C" void kernel_launch(void* const* d_in, const int* in_sizes, int n_in,
                              void* d_out, int out_size, void* d_ws, size_t ws_size,
                              hipStream_t stream)
{
    const float* X    = (const float*)d_in[0];   // hidden_frames (nImg, 1024)
    const float* W_hc = (const float*)d_in[1];   // (1536, 512)
    const float* b_hc = (const float*)d_in[2];
    const float* W_wg = (const float*)d_in[3];   // (1536, 1)
    const float* b_wg = (const float*)d_in[4];
    const float* W_wp = (const float*)d_in[5];   // (1536, 100)
    const float* b_wp = (const float*)d_in[6];
    const float* W_rg = (const float*)d_in[7];   // (1536, 1)
    const float* b_rg = (const float*)d_in[8];
    const float* W_rp = (const float*)d_in[9];   // (1536, 100)
    const float* b_rp = (const float*)d_in[10];
    const float* Wxh  = (const float*)d_in[11];  // (1024, 512)
    const float* Wrh  = (const float*)d_in[12];  // (512, 512)
    const float* Whh  = (const float*)d_in[13];  // (512, 512)
    const float* bh   = (const float*)d_in[14];
    const int nImg = in_sizes[0] / DI;           // 4096 (d_in[15] is a device scalar)

    // Workspace: precomputed x-parts (~19.2 MB fp32)
    float* ws   = (float*)d_ws;
    float* P_xh = ws;                                   // nImg*512
    float* P_hc = P_xh + (size_t)nImg * DH;             // nImg*512
    float* P_wp = P_hc + (size_t)nImg * DH;             // nImg*100
    float* P_rp = P_wp + (size_t)nImg * DM;             // nImg*100
    float* P_wg = P_rp + (size_t)nImg * DM;             // nImg
    float* P_rg = P_wg + nImg;                          // nImg

    const int gx = (nImg + 127) / 128;                  // 128 rows per block
    wmma_gemm_f32<false><<<dim3(gx, DH / 16), 256, 0, stream>>>(X, Wxh,  P_xh, nImg, DI, DH, DI, DH, DH);
    wmma_gemm_f32<false><<<dim3(gx, DH / 16), 256, 0, stream>>>(X, W_hc, P_hc, nImg, DI, DH, DI, DH, DH);
    wmma_gemm_f32<true ><<<dim3(gx, (DM + 15) / 16), 256, 0, stream>>>(X, W_wp, P_wp, nImg, DI, DM, DI, DM, DM);
    wmma_gemm_f32<true ><<<dim3(gx, (DM + 15) / 16), 256, 0, stream>>>(X, W_rp, P_rp, nImg, DI, DM, DI, DM, DM);
    gates_kernel<<<nImg, 256, 0, stream>>>(X, W_wg, W_rg, P_wg, P_rg);

    // LDS: mem(51200) + h/r/c(3*512) + ar/aw/red(3*128) + p0/p1(2*1024) + scal(8)
    const size_t smemBytes =
        (size_t)(DM * DH + 3 * DH + 3 * 128 + 2 * 1024 + 8) * sizeof(float);   // ~216 KB < 320 KB
    scan_kernel<<<1, 1024, smemBytes, stream>>>(
        P_xh, P_hc, P_wp, P_rp, P_wg, P_rg,
        W_hc, b_hc, W_wg, b_wg, W_wp, b_wp, W_rg, b_rg, W_rp, b_rp,
        Wrh, Whh, bh, (float*)d_out, nImg);
}